// HopfRNNLayerRadial_80221399154863
// MI455X (gfx1250) — compile-verified
//
#include <hip/hip_runtime.h>

// Hopf RNN (radial tanh), fused input projection + recurrence.
// x:[256,1024,64] f32, A:[64,64], B:[64,64]  ->  y:[256,1024,64] f32
//
// 16 blocks (16 batch rows) x 128 threads (4 waves, one 16-col N-tile each).
// Recurrence carried as s_t = 2*z_t:  s_t = x_t@A + s_{t-1}@(0.5*B),
// y_t = tanh(s_t). Both GEMMs use V_WMMA_F32_16X16X4_F32 (f32, matches
// reference precision). Weights (0.5*B folded) live in registers in B-operand
// layout; s is exchanged across waves via double-buffered, bank-conflict-
// padded LDS.
//
// Critical path per step: s ds_loads -> 8-deep zB WMMA chains -> tanh/add ->
// s ds_stores -> s_wait_dscnt 0 -> s_barrier_signal. The 16 xA WMMAs for
// step t+1 run between signal and wait (barrier shadow). No storecnt waits
// in the loop: LDS-only ordering via explicit s_wait_dscnt instead of a
// workgroup fence.

typedef float v2f __attribute__((ext_vector_type(2)));
typedef float v8f __attribute__((ext_vector_type(8)));

#define T_STEPS 1024
#define FEAT    64
#define LDSROW  66   // padded row stride (floats): banks (66*m + k) % 64 unique

__device__ __forceinline__ v8f wmma4(v2f a, v2f b, v8f c) {
    // (neg_a, A, neg_b, B, c_mod, C, reuse_a, reuse_b)
    return __builtin_amdgcn_wmma_f32_16x16x4_f32(false, a, false, b,
                                                 (short)0, c, false, false);
}

__device__ __forceinline__ float act_tanh(float x) {
#if __has_builtin(__builtin_amdgcn_tanhf)
    return __builtin_amdgcn_tanhf(x);     // V_TANH_F32 (single TRANS op)
#else
    return tanhf(x);
#endif
}

__global__ void __launch_bounds__(128)
hopf_rnn_wmma_f32(const float* __restrict__ x,
                  const float* __restrict__ Amat,
                  const float* __restrict__ Bmat,
                  float* __restrict__ y)
{
    __shared__ float zbuf[2][16 * LDSROW];

    const int lane = threadIdx.x & 31;
    const int wave = threadIdx.x >> 5;          // 0..3  -> N-tile
    const int b0   = blockIdx.x * 16;           // batch-tile base row

    const int m    = lane & 15;                 // M row (A-op) / N col (B,C)
    const int hi   = lane >> 4;                 // lane half
    const int ncol = (wave << 4) + m;           // this wave's output column

    // ---- Weights in registers, B-operand layout ----------------------------
    // chunk c covers K rows [4c,4c+4); v2f elem j holds k = 4c + 2*hi + j.
    // wB is pre-scaled by 0.5 (recurrence carried as s = 2*z).
    v2f wA[16], wB[16];
#pragma unroll
    for (int c = 0; c < 16; ++c) {
        const int k0 = (c << 2) + (hi << 1);
        wA[c].x = Amat[(k0 + 0) * FEAT + ncol];
        wA[c].y = Amat[(k0 + 1) * FEAT + ncol];
        wB[c].x = 0.5f * Bmat[(k0 + 0) * FEAT + ncol];
        wB[c].y = 0.5f * Bmat[(k0 + 1) * FEAT + ncol];
    }

    // ---- Per-lane base pointers --------------------------------------------
    // A-operand x chunk c, step t: x[(b0+m), t, 4c + 2*hi + {0,1}] (b64 load)
    const float* xlane = x + ((size_t)(b0 + m) * T_STEPS) * FEAT + (hi << 1);
    // C-layout rows: vgpr i -> batch row b0 + i + 8*hi, column ncol
    float* ylane = y + ((size_t)(b0 + (hi << 3)) * T_STEPS) * FEAT + ncol;

    // ---- State s = 2*z (A-operand layout), starts at zero ------------------
    v2f sA[16];
#pragma unroll
    for (int c = 0; c < 16; ++c) sA[c] = (v2f){0.0f, 0.0f};

    // Prefetch x for t=0 and pre-compute its xA accumulation
    v2f xcur[16];
#pragma unroll
    for (int c = 0; c < 16; ++c)
        xcur[c] = *(const v2f*)(xlane + (c << 2));

    v8f ax0 = {0.f,0.f,0.f,0.f,0.f,0.f,0.f,0.f};
    v8f ax1 = {0.f,0.f,0.f,0.f,0.f,0.f,0.f,0.f};
#pragma unroll
    for (int c = 0; c < 16; c += 2) {
        ax0 = wmma4(xcur[c    ], wA[c    ], ax0);
        ax1 = wmma4(xcur[c + 1], wA[c + 1], ax1);
    }

    for (int t = 0; t < T_STEPS; ++t) {
        // Prefetch x for next step (clamped; last-iter reload is harmless)
        const int tn = (t + 1 < T_STEPS) ? (t + 1) : t;
        v2f xnxt[16];
#pragma unroll
        for (int c = 0; c < 16; ++c)
            xnxt[c] = *(const v2f*)(xlane + (size_t)tn * FEAT + (c << 2));

        // s = (precomputed x_t@A) + s_prev@(0.5B)  (two interleaved chains)
        v8f acc0 = ax0, acc1 = ax1;
#pragma unroll
        for (int c = 0; c < 16; c += 2) {
            acc0 = wmma4(sA[c    ], wB[c    ], acc0);
            acc1 = wmma4(sA[c + 1], wB[c + 1], acc1);
        }

        const int buf = t & 1;
        float* zrow = &zbuf[buf][0];

        // Publish s (no scaling needed) and emit y_t = tanh(s)
#pragma unroll
        for (int i = 0; i < 8; ++i) {
            const float a = acc0[i] + acc1[i];
            zrow[(i + (hi << 3)) * LDSROW + ncol] = a;          // C layout row
            __builtin_nontemporal_store(act_tanh(a),
                ylane + (size_t)(i * T_STEPS + t) * FEAT);
        }

        // ---- Split barrier: LDS-only ordering (no storecnt wait) -----------
        asm volatile("s_wait_dscnt 0x0" ::: "memory");   // s visible in LDS
        __builtin_amdgcn_s_barrier_signal(-1);

        // Next step's input projection (independent of s) — barrier shadow
        ax0 = (v8f){0.f,0.f,0.f,0.f,0.f,0.f,0.f,0.f};
        ax1 = (v8f){0.f,0.f,0.f,0.f,0.f,0.f,0.f,0.f};
#pragma unroll
        for (int c = 0; c < 16; c += 2) {
            ax0 = wmma4(xnxt[c    ], wA[c    ], ax0);
            ax1 = wmma4(xnxt[c + 1], wA[c + 1], ax1);
        }

        __builtin_amdgcn_s_barrier_wait(-1);
        asm volatile("" ::: "memory");                   // no hoist of s loads

        // Reload s in A-operand layout (b64, padded rows -> conflict-free)
#pragma unroll
        for (int c = 0; c < 16; ++c)
            sA[c] = *(const v2f*)&zrow[m * LDSROW + (c << 2) + (hi << 1)];
    }
}

extern "C" void kernel_launch(void* const* d_in, const int* in_sizes, int n_in,
                              void* d_out, int out_size, void* d_ws, size_t ws_size,
                              hipStream_t stream) {
    const float* x = (const float*)d_in[0];
    const float* A = (const float*)d_in[1];
    const float* B = (const float*)d_in[2];
    float* out = (float*)d_out;

    const int batch = in_sizes[0] / (T_STEPS * FEAT);   // 256
    const int tiles = batch / 16;                       // 16 blocks

    hipLaunchKernelGGL(hopf_rnn_wmma_f32, dim3(tiles), dim3(128), 0, stream,
                       x, A, B, out);
}